// RNPDecoder_72516227826255
// MI455X (gfx1250) — compile-verified
//
#include <hip/hip_runtime.h>

// ---------------------------------------------------------------------------
// RNP decoder for MI455X (gfx1250).
//  Stage A: hypernet backbones as f32 WMMA GEMMs (v_wmma_f32_16x16x4_f32),
//           fused bias + relu. K templated (fully unrolled), 4 M-tiles per
//           wave share each B fragment (4x less W traffic, L2-resident).
//  Stage B: 1 block per sample, sequential GRU/dense hypernet recursion in LDS.
//           x input is provably unused by the reference (dead data flow).
// ---------------------------------------------------------------------------

typedef __attribute__((ext_vector_type(2))) float v2f;
typedef __attribute__((ext_vector_type(8))) float v8f;

#define HH 28
#define WW 28
#define NPIX 784

// ---------------- Stage A: WMMA f32 GEMM, C = act(A[MxK] @ W[KxN] + bias) ---
// One wave computes MT vertically-stacked 16x16 tiles (rows m0..m0+16*MT-1,
// cols n0..n0+15), so the B fragment is loaded once per K-step for MT WMMAs.
// Out-of-range columns are handled by clamping the load address: a garbage
// B column only affects the same (unstored) C column, so no predication and
// no exec-mask churn around the vmem ops.
template <int K, int MT>
__global__ __launch_bounds__(256) void gemm_wmma_f32(
    const float* __restrict__ A, const float* __restrict__ W,
    const float* __restrict__ bias, float* __restrict__ C,
    int M, int N, int relu)
{
  const int wave   = (blockIdx.x * blockDim.x + threadIdx.x) >> 5;
  const int lane   = threadIdx.x & 31;
  const int tilesN = (N + 15) >> 4;
  const int tm = wave / tilesN;
  const int tn = wave - tm * tilesN;
  const int m0 = tm * (16 * MT);
  if (m0 >= M) return;                 // also covers wave >= gridWaves
  const int n0   = tn << 4;
  const int lrow = lane & 15;
  const int hi   = lane >> 4;          // 0: K pair {0,1}, 1: K pair {2,3}

  const int ccol  = n0 + lrow;                 // B/C column for this lane
  const int ccolc = (ccol < N) ? ccol : N - 1; // clamped load address

  v8f acc[MT] = {};

#pragma unroll
  for (int k = 0; k < K; k += 4) {
    const int kk = k + 2 * hi;
    v2f b;
    b[0] = W[(size_t)(kk    ) * N + ccolc];
    b[1] = W[(size_t)(kk + 1) * N + ccolc];
#pragma unroll
    for (int t = 0; t < MT; ++t) {
      const int arow = m0 + t * 16 + lrow;     // M % (16*MT) == 0 here
      v2f a;
      a[0] = A[(size_t)arow * K + kk];
      a[1] = A[(size_t)arow * K + kk + 1];
      acc[t] = __builtin_amdgcn_wmma_f32_16x16x4_f32(
          /*neg_a=*/false, a, /*neg_b=*/false, b,
          /*c_mod=*/(short)0, acc[t], /*reuse_a=*/false, /*reuse_b=*/false);
    }
  }

  if (ccol < N) {
    const float bb = bias[ccol];
#pragma unroll
    for (int t = 0; t < MT; ++t) {
#pragma unroll
      for (int r = 0; r < 8; ++r) {
        float v = acc[t][r] + bb;
        if (relu) v = fmaxf(v, 0.0f);
        C[(size_t)(m0 + t * 16 + r + 8 * hi) * N + ccol] = v;  // row = r + 8*hi
      }
    }
  }
}

// ---------------- Stage B helpers (block-cooperative, smem vectors) ---------
__device__ __forceinline__ void dense_sm(const float* __restrict__ p,
                                         const float* in, float* outv,
                                         int din, int dout, bool do_relu)
{
  __syncthreads();
  const float* bp = p + (size_t)din * dout;
  for (int o = threadIdx.x; o < dout; o += 256) {
    float acc = bp[o];
    for (int i = 0; i < din; ++i) acc = fmaf(in[i], p[(size_t)i * dout + o], acc);
    outv[o] = do_relu ? fmaxf(acc, 0.0f) : acc;
  }
  __syncthreads();
}

__device__ __forceinline__ void gru_sm(const float* __restrict__ rnnp,
                                       const float* e, float* h,
                                       float* gxv, float* zg, float* rghv)
{
  __syncthreads();
  const float* k  = rnnp;               // 32 x 96
  const float* rk = rnnp + 32 * 96;     // 32 x 96
  const float* bb = rnnp + 2 * 32 * 96; // 96
  const int t = threadIdx.x;
  if (t < 96) {
    float acc = bb[t];
    for (int i = 0; i < 32; ++i) acc = fmaf(e[i], k[i * 96 + t], acc);
    gxv[t] = acc;
  }
  __syncthreads();
  if (t < 64) {
    const int o = t & 31, w = t >> 5;   // w==0: update gate, w==1: reset gate
    float acc = gxv[w * 32 + o];
    for (int i = 0; i < 32; ++i) acc = fmaf(h[i], rk[i * 96 + w * 32 + o], acc);
    const float s = 1.0f / (1.0f + __expf(-acc));
    if (w == 0) zg[o] = s; else rghv[o] = s * h[o];
  }
  __syncthreads();
  if (t < 32) {
    float acc = gxv[64 + t];
    for (int i = 0; i < 32; ++i) acc = fmaf(rghv[i], rk[i * 96 + 64 + t], acc);
    const float hh = tanhf(acc);
    const float z = zg[t];
    h[t] = z * h[t] + (1.0f - z) * hh;
  }
  __syncthreads();
}

// out-of-bounds reads are zero (matches the reference's validity mask)
__device__ __forceinline__ void sample_add(const float* img, const float* a, float* dst)
{
  __syncthreads();
  const float sx = a[0] + 1.0f, sy = a[1] + 1.0f, tx = a[2], ty = a[3];
  for (int idx = threadIdx.x; idx < NPIX; idx += 256) {
    const int iy = idx / WW, ix = idx - iy * WW;
    const float gxc = -1.0f + 2.0f * (float)ix * (1.0f / 27.0f);
    const float gyc = -1.0f + 2.0f * (float)iy * (1.0f / 27.0f);
    const float px = (sx * gxc + tx + 1.0f) * 0.5f * 27.0f;
    const float py = (sy * gyc + ty + 1.0f) * 0.5f * 27.0f;
    const float x0f = floorf(px), y0f = floorf(py);
    const float wx = px - x0f, wy = py - y0f;
    const int x0 = (int)x0f, y0 = (int)y0f;
    float v = 0.0f;
#pragma unroll
    for (int dy = 0; dy < 2; ++dy)
#pragma unroll
      for (int dx = 0; dx < 2; ++dx) {
        const int yy = y0 + dy, xx = x0 + dx;
        const float wgt = (dx ? wx : 1.0f - wx) * (dy ? wy : 1.0f - wy);
        if (yy >= 0 && yy < HH && xx >= 0 && xx < WW) v += img[yy * WW + xx] * wgt;
      }
    dst[idx] += v;
  }
  __syncthreads();
}

// one full hyper step (enc1 -> enc2 -> GRU -> [dec1 -> dec2])
__device__ __forceinline__ void hyper_step(
    const float* __restrict__ base,
    int o_enc1, int o_enc2, int o_dec1, int o_dec2, int o_rnn, int dec_out,
    float* h, const float* z0v, const float* za0v,
    float* xvv, float* e1v, float* e2v, float* gxv, float* zg, float* rghv,
    float* d1v, float* decv, bool do_dec)
{
  __syncthreads();
  if (threadIdx.x < 32) {
    xvv[threadIdx.x]      = z0v[threadIdx.x];
    xvv[32 + threadIdx.x] = za0v[threadIdx.x];
  }
  dense_sm(base + o_enc1, xvv, e1v, 64, 32, true);
  dense_sm(base + o_enc2, e1v, e2v, 32, 32, false);
  gru_sm(base + o_rnn, e2v, h, gxv, zg, rghv);
  if (do_dec) {
    dense_sm(base + o_dec1, h, d1v, 32, 32, true);
    dense_sm(base + o_dec2, d1v, decv, 32, dec_out, false);
  }
}

// ---------------- Stage B: 1 block == 1 sample ------------------------------
__global__ __launch_bounds__(256) void rnp_kernel(
    const float* __restrict__ S, const float* __restrict__ P,
    float* __restrict__ out)
{
  const int b = blockIdx.x;
  const int t = threadIdx.x;
  const float* Sb = S + (size_t)b * 36336;
  const float* Pb = P + (size_t)b * 10596;
  // S offsets: enc1@0(2080) enc2@2080(1056) dec1@3136(1056) dec2@4192(25872) rnn@30064(6240) z0@36304
  // P offsets: enc1@0 enc2@2080 dec1@3136 dec2@4192(132) rnn@4324 za0@10564

  __shared__ float xv[64], e1[32], e2[32], gx[96], zgate[32], rgh[32];
  __shared__ float ss[32], sp[32];
  __shared__ float z0[32], za0[32];      // current step inputs
  __shared__ float z0on[32], za0on[32];  // pending outer-next snapshots
  __shared__ float d1[32];
  __shared__ float xhat[NPIX];
  __shared__ float avec[4], aouter[4];
  __shared__ float oinner[NPIX], oacc[NPIX];

  if (t < 32) { ss[t] = 0.0f; sp[t] = 0.0f; z0[t] = Sb[36304 + t]; za0[t] = Pb[10564 + t]; }
  for (int i = t; i < NPIX; i += 256) oacc[i] = 0.0f;
  __syncthreads();

  for (int to = 0; to < 4; ++to) {
    // outer s step: dec output (x_hat) is dead at level>0 -> skip dec1/dec2
    hyper_step(Sb, 0, 2080, 3136, 4192, 30064, NPIX, ss, z0, za0,
               xv, e1, e2, gx, zgate, rgh, d1, xhat, false);
    if (t < 32) z0on[t] = ss[t];                      // nz snapshot
    // outer p step: need transform params a
    hyper_step(Pb, 0, 2080, 3136, 4192, 4324, 4, sp, z0, za0,
               xv, e1, e2, gx, zgate, rgh, d1, aouter, true);
    if (t < 32) za0on[t] = sp[t];                     // nza snapshot
    __syncthreads();

    // inner level-0 recursion: z0/za0 restart at init, out_inner = 0
    if (t < 32) { z0[t] = Sb[36304 + t]; za0[t] = Pb[10564 + t]; }
    for (int i = t; i < NPIX; i += 256) oinner[i] = 0.0f;
    __syncthreads();

    for (int ti = 0; ti < 4; ++ti) {
      hyper_step(Sb, 0, 2080, 3136, 4192, 30064, NPIX, ss, z0, za0,
                 xv, e1, e2, gx, zgate, rgh, d1, xhat, true);
      hyper_step(Pb, 0, 2080, 3136, 4192, 4324, 4, sp, z0, za0,
                 xv, e1, e2, gx, zgate, rgh, d1, avec, true);
      __syncthreads();
      if (t < 32) { z0[t] = ss[t]; za0[t] = sp[t]; }  // (nz, nza)
      sample_add(xhat, avec, oinner);
    }
    sample_add(oinner, aouter, oacc);
    __syncthreads();
    if (t < 32) { z0[t] = z0on[t]; za0[t] = za0on[t]; }
    __syncthreads();
  }

  for (int i = t; i < NPIX; i += 256) out[(size_t)b * NPIX + i] = oacc[i];
}

// ---------------------------------------------------------------------------
extern "C" void kernel_launch(void* const* d_in, const int* in_sizes, int n_in,
                              void* d_out, int out_size, void* d_ws, size_t ws_size,
                              hipStream_t stream) {
  (void)in_sizes; (void)n_in; (void)out_size; (void)ws_size;
  // inputs: x(unused), z, s_w1,s_b1,s_w2,s_b2,s_w3,s_b3, p_w1,p_b1,p_w2,p_b2,p_w3,p_b3
  const float* z    = (const float*)d_in[1];
  const float* s_w1 = (const float*)d_in[2];
  const float* s_b1 = (const float*)d_in[3];
  const float* s_w2 = (const float*)d_in[4];
  const float* s_b2 = (const float*)d_in[5];
  const float* s_w3 = (const float*)d_in[6];
  const float* s_b3 = (const float*)d_in[7];
  const float* p_w1 = (const float*)d_in[8];
  const float* p_b1 = (const float*)d_in[9];
  const float* p_w2 = (const float*)d_in[10];
  const float* p_b2 = (const float*)d_in[11];
  const float* p_w3 = (const float*)d_in[12];
  const float* p_b3 = (const float*)d_in[13];

  const size_t Bsz = 1024, SOUT = 36336, POUT = 10596;
  float* ws   = (float*)d_ws;           // needs ~193 MB: S + P + tiny h buffers
  float* Sbuf = ws;
  float* Pbuf = Sbuf + Bsz * SOUT;
  float* h1s  = Pbuf + Bsz * POUT;
  float* h2s  = h1s + Bsz * 64;
  float* h1p  = h2s + Bsz * 64;
  float* h2p  = h1p + Bsz * 64;

  constexpr int MT = 4;                  // M-tiles per wave (rows per wave = 64)
  auto blocks_for = [](int M, int N) {
    const long tilesN = (N + 15) / 16;
    const long waves  = (long)(M / (16 * MT)) * tilesN;
    return (int)((waves + 7) / 8);       // 8 wave32 per 256-thread block
  };

  // s-chain backbone
  gemm_wmma_f32<32, MT><<<blocks_for(1024, 64), 256, 0, stream>>>(
      z, s_w1, s_b1, h1s, 1024, 64, 1);
  gemm_wmma_f32<64, MT><<<blocks_for(1024, 64), 256, 0, stream>>>(
      h1s, s_w2, s_b2, h2s, 1024, 64, 1);
  gemm_wmma_f32<64, MT><<<blocks_for(1024, 36336), 256, 0, stream>>>(
      h2s, s_w3, s_b3, Sbuf, 1024, 36336, 0);
  // p-chain backbone
  gemm_wmma_f32<32, MT><<<blocks_for(1024, 64), 256, 0, stream>>>(
      z, p_w1, p_b1, h1p, 1024, 64, 1);
  gemm_wmma_f32<64, MT><<<blocks_for(1024, 64), 256, 0, stream>>>(
      h1p, p_w2, p_b2, h2p, 1024, 64, 1);
  gemm_wmma_f32<64, MT><<<blocks_for(1024, 10596), 256, 0, stream>>>(
      h2p, p_w3, p_b3, Pbuf, 1024, 10596, 0);

  // sequential per-sample hypernet recursion
  rnp_kernel<<<1024, 256, 0, stream>>>(Sbuf, Pbuf, (float*)d_out);
}